// EGCNGuard_69389491634485
// MI455X (gfx1250) — compile-verified
//
#include <hip/hip_runtime.h>

// ---------------------------------------------------------------------------
// EGCNGuard: 3-layer GCN with cosine-similarity edge pruning, fp32 end-to-end.
// Dense GEMMs use V_WMMA_F32_16X16X4_F32 (exact fp32 matrix path on CDNA5).
// Graph aggregation is wave-per-edge gather + global f32 atomic scatter; the
// whole feature working set (<40MB) is L2-resident (192MB L2), so this is an
// L2-bandwidth-bound kernel sequence.
// ---------------------------------------------------------------------------

#define N_NODES 50000
#define N_EDGES 800000
#define THRESH  0.1f

typedef __attribute__((ext_vector_type(2))) float v2f;
typedef __attribute__((ext_vector_type(8))) float v8f;

__device__ __forceinline__ float wave_sum(float s) {
  #pragma unroll
  for (int off = 16; off > 0; off >>= 1) s += __shfl_xor(s, off, 32);
  return s;
}
__device__ __forceinline__ float wave_max(float s) {
  #pragma unroll
  for (int off = 16; off > 0; off >>= 1) s = fmaxf(s, __shfl_xor(s, off, 32));
  return s;
}

// ---- 1) per-node L2 norm (clamped at 1e-8) + deg := 1.0 (self loop) --------
__global__ void k_norm_deg(const float* __restrict__ X, int C,
                           float* __restrict__ nrm, float* __restrict__ deg) {
  int n = blockIdx.x * (blockDim.x >> 5) + (threadIdx.x >> 5);
  if (n >= N_NODES) return;
  int lane = threadIdx.x & 31;
  const float* xr = X + (size_t)n * C;
  float s = 0.f;
  for (int i = lane; i < C; i += 32) { float v = xr[i]; s += v * v; }
  s = wave_sum(s);
  if (lane == 0) {
    nrm[n] = fmaxf(sqrtf(s), 1e-8f);
    deg[n] = 1.0f;
  }
}

// ---- 2) H[N,Cout] = X[N,K] @ W[K,Cout] via fp32 WMMA 16x16x4 ---------------
// One wave computes one 16x16 output tile; W staged in LDS padded to CoutPad.
// A layout (16x4 f32): lane L: m=L%16, VGPR0/1 = K = (L<16 ? 0,1 : 2,3)
// B layout (4x16 f32): lane L: n=L%16, VGPR0/1 = K = (L<16 ? 0,1 : 2,3)
// C/D (16x16 f32):     lane L: n=L%16, VGPR r -> M = r + 8*(L>=16)
__global__ void k_gemm_wmma_f32(const float* __restrict__ X,
                                const float* __restrict__ W,
                                float* __restrict__ H,
                                int K, int Cout, int CoutPad) {
  __shared__ float Wlds[8192];                    // up to 128x64 fp32 = 32KB
  int tid = threadIdx.x;
  int total = K * CoutPad;
  for (int idx = tid; idx < total; idx += blockDim.x) {
    int k = idx / CoutPad, n = idx % CoutPad;
    Wlds[idx] = (n < Cout) ? W[k * Cout + n] : 0.0f;
  }
  __syncthreads();

  int tilesN = CoutPad >> 4;
  int tilesM = N_NODES >> 4;                      // 3125, exact
  int wid  = tid >> 5;
  int lane = tid & 31;
  int tile = blockIdx.x * (blockDim.x >> 5) + wid;
  if (tile >= tilesM * tilesN) return;            // wave-uniform -> EXEC all-1s

  int m0 = (tile / tilesN) << 4;
  int n0 = (tile % tilesN) << 4;
  int mn = lane & 15;
  int kh = (lane >> 4) << 1;                      // 0 or 2

  v8f acc = {};
  const float* xrow = X + (size_t)(m0 + mn) * K + kh;
  for (int k = 0; k < K; k += 4) {
    v2f a, b;
    a.x = xrow[k];
    a.y = xrow[k + 1];
    b.x = Wlds[(k + kh) * CoutPad + n0 + mn];
    b.y = Wlds[(k + kh + 1) * CoutPad + n0 + mn];
    acc = __builtin_amdgcn_wmma_f32_16x16x4_f32(
        /*neg_a=*/false, a, /*neg_b=*/false, b,
        /*c_mod=*/(short)0, acc, /*reuse_a=*/false, /*reuse_b=*/false);
  }

  int rbase = (lane >> 4) << 3;
  int n = n0 + mn;
  if (n < Cout) {
    #pragma unroll
    for (int r = 0; r < 8; ++r)
      H[(size_t)(m0 + rbase + r) * Cout + n] = acc[r];
  }
}

// ---- 3) wave-per-edge cosine similarity mask + degree accumulation ---------
__global__ void k_edge_mask(const float* __restrict__ X, int C,
                            const float* __restrict__ nrm,
                            const int* __restrict__ row, const int* __restrict__ col,
                            unsigned char* __restrict__ mask,
                            float* __restrict__ deg) {
  int e = blockIdx.x * (blockDim.x >> 5) + (threadIdx.x >> 5);
  if (e >= N_EDGES) return;
  int lane = threadIdx.x & 31;
  int r = row[e], c = col[e];
  const float* xr = X + (size_t)r * C;
  const float* xc = X + (size_t)c * C;
  float s = 0.f;
  for (int i = lane; i < C; i += 32) s += xr[i] * xc[i];
  s = wave_sum(s);
  if (lane == 0) {
    float sim = s / (nrm[r] * nrm[c]);
    bool m = (sim >= THRESH) || (r == c);
    mask[e] = m ? 1u : 0u;
    if (m) atomicAdd(&deg[c], 1.0f);
  }
}

// ---- 4) dinv = deg > 0 ? rsqrt(deg) : 0 ------------------------------------
__global__ void k_dinv(const float* __restrict__ deg, float* __restrict__ dinv) {
  int i = blockIdx.x * blockDim.x + threadIdx.x;
  if (i >= N_NODES) return;
  float d = deg[i];
  dinv[i] = (d > 0.f) ? __frsqrt_rn(d) : 0.f;
}

// ---- 5) out[i,c] = b[c] + dinv[i]^2 * h[i,c]  (self-loop + bias) -----------
__global__ void k_init_out(const float* __restrict__ H, const float* __restrict__ bias,
                           const float* __restrict__ dinv, float* __restrict__ out,
                           int Cout) {
  int t = blockIdx.x * blockDim.x + threadIdx.x;
  if (t >= N_NODES * Cout) return;
  int i = t / Cout, c = t - i * Cout;
  float di = dinv[i];
  out[t] = bias[c] + di * di * H[t];
}

// ---- 6) scatter: out[col] += dinv[row]*dinv[col] * h[row] ------------------
// 64 lanes per edge (guarded for Cout=40); atomic f32 adds into L2-resident out.
__global__ void k_scatter(const float* __restrict__ H,
                          const int* __restrict__ row, const int* __restrict__ col,
                          const unsigned char* __restrict__ mask,
                          const float* __restrict__ dinv,
                          float* __restrict__ out, int Cout) {
  long long t = (long long)blockIdx.x * blockDim.x + threadIdx.x;
  int e = (int)(t >> 6);
  int c = (int)(t & 63);
  if (e >= N_EDGES || c >= Cout) return;
  if (!mask[e]) return;
  int r = row[e], cc = col[e];
  float coef = dinv[r] * dinv[cc];
  atomicAdd(&out[(size_t)cc * Cout + c], H[(size_t)r * Cout + c] * coef);
}

// ---- 7a) ReLU --------------------------------------------------------------
__global__ void k_relu(float* __restrict__ x, int total) {
  int t = blockIdx.x * blockDim.x + threadIdx.x;
  if (t < total) x[t] = fmaxf(x[t], 0.f);
}

// ---- 7b) wave-per-node log_softmax over 40 classes -------------------------
__global__ void k_log_softmax(const float* __restrict__ in, float* __restrict__ out) {
  int n = blockIdx.x * (blockDim.x >> 5) + (threadIdx.x >> 5);
  if (n >= N_NODES) return;
  int lane = threadIdx.x & 31;
  const float* r = in + (size_t)n * 40;
  float v0 = r[lane];                               // lanes 0..31 -> c=0..31
  float v1 = (lane < 8) ? r[lane + 32] : -3.4e38f;  // lanes 0..7  -> c=32..39
  float mx = wave_max(fmaxf(v0, v1));
  float se = __expf(v0 - mx) + ((lane < 8) ? __expf(v1 - mx) : 0.f);
  se = wave_sum(se);
  float lse = mx + __logf(se);
  float* w = out + (size_t)n * 40;
  w[lane] = v0 - lse;
  if (lane < 8) w[lane + 32] = v1 - lse;
}

// ---------------------------------------------------------------------------
static void run_layer(const float* xin, int Cin,
                      const float* W, const float* bias,
                      int Cout, int CoutPad,
                      float* H, float* outb,
                      float* nrm, float* deg, float* dinv, unsigned char* mask,
                      const int* row, const int* col,
                      bool do_relu, hipStream_t stream) {
  const int WPB = 8;  // waves per 256-thread block
  int nodeWaveBlocks = (N_NODES + WPB - 1) / WPB;
  int edgeWaveBlocks = (N_EDGES + WPB - 1) / WPB;

  k_norm_deg<<<nodeWaveBlocks, 256, 0, stream>>>(xin, Cin, nrm, deg);

  int tiles = (N_NODES >> 4) * (CoutPad >> 4);
  k_gemm_wmma_f32<<<(tiles + WPB - 1) / WPB, 256, 0, stream>>>(xin, W, H, Cin, Cout, CoutPad);

  k_edge_mask<<<edgeWaveBlocks, 256, 0, stream>>>(xin, Cin, nrm, row, col, mask, deg);

  k_dinv<<<(N_NODES + 255) / 256, 256, 0, stream>>>(deg, dinv);

  int totalOut = N_NODES * Cout;
  k_init_out<<<(totalOut + 255) / 256, 256, 0, stream>>>(H, bias, dinv, outb, Cout);

  long long scatterThreads = (long long)N_EDGES * 64;
  k_scatter<<<(int)((scatterThreads + 255) / 256), 256, 0, stream>>>(
      H, row, col, mask, dinv, outb, Cout);

  if (do_relu)
    k_relu<<<(totalOut + 255) / 256, 256, 0, stream>>>(outb, totalOut);
}

extern "C" void kernel_launch(void* const* d_in, const int* in_sizes, int n_in,
                              void* d_out, int out_size, void* d_ws, size_t ws_size,
                              hipStream_t stream) {
  const float* x  = (const float*)d_in[0];   // [N,128]
  const float* W1 = (const float*)d_in[1];   // [128,64]
  const float* b1 = (const float*)d_in[2];   // [64]
  const float* W2 = (const float*)d_in[3];   // [64,64]
  const float* b2 = (const float*)d_in[4];   // [64]
  const float* W3 = (const float*)d_in[5];   // [64,40]
  const float* b3 = (const float*)d_in[6];   // [40]
  const int*   ei = (const int*)d_in[7];     // [2,E]
  const int* row = ei;
  const int* col = ei + N_EDGES;
  float* out = (float*)d_out;                // [N,40]

  // workspace carve-up (~40MB)
  float* H    = (float*)d_ws;                       // [N,64]
  float* xA   = H  + (size_t)N_NODES * 64;          // [N,64]
  float* xB   = xA + (size_t)N_NODES * 64;          // [N,64]
  float* nrm  = xB + (size_t)N_NODES * 64;          // [N]
  float* deg  = nrm + N_NODES;                      // [N]
  float* dinv = deg + N_NODES;                      // [N]
  unsigned char* mask = (unsigned char*)(dinv + N_NODES);  // [E]

  // layer 1: x[N,128] -> xA[N,64], relu
  run_layer(x,  128, W1, b1, 64, 64, H, xA, nrm, deg, dinv, mask, row, col, true,  stream);
  // layer 2: xA[N,64] -> xB[N,64], relu
  run_layer(xA,  64, W2, b2, 64, 64, H, xB, nrm, deg, dinv, mask, row, col, true,  stream);
  // layer 3: xB[N,64] -> xA[N,40] (reuse xA as [N,40] staging), no relu
  run_layer(xB,  64, W3, b3, 40, 48, H, xA, nrm, deg, dinv, mask, row, col, false, stream);

  // log_softmax over 40 classes -> d_out
  k_log_softmax<<<(N_NODES + 7) / 8, 256, 0, stream>>>(xA, out);
}